// PairwiseCrossSimilarity_21921513079475
// MI455X (gfx1250) — compile-verified
//
#include <hip/hip_runtime.h>
#include <hip/hip_bf16.h>

typedef __attribute__((ext_vector_type(2))) float v2f;
typedef __attribute__((ext_vector_type(8))) float v8f;

#define NUMG   32
#define LROWS  128   // num_top * cobj
#define BAG    300
#define DDIM   256
#define HDIM   256
#define POOL   1200  // cobj * bag

// ---------------------------------------------------------------------------
// Kernel 1: fused gather + GEMM via f32 WMMA.
// One wave per 16x16 tile of H[n][side] = gather(orig)[128,256] @ W[256,256].
// 8192 tiles total -> 1024 blocks x 8 waves. K-loop: 64 x v_wmma_f32_16x16x4.
// b1 is folded into side-1 (it lives inside the relu of the next stage).
// ---------------------------------------------------------------------------
__global__ __launch_bounds__(256) void gemm_gather_wmma(
    const float* __restrict__ orig, const int* __restrict__ ind0,
    const int* __restrict__ ind1, const float* __restrict__ W1a,
    const float* __restrict__ W1b, const float* __restrict__ b1,
    float* __restrict__ Hbuf)
{
  const int wave = (blockIdx.x * blockDim.x + threadIdx.x) >> 5;
  const int lane = threadIdx.x & 31;
  const int lo = lane & 15, hi = lane >> 4;

  const int ht   = wave & 15;        // 16 tiles along H
  const int mt   = (wave >> 4) & 7;  // 8 tiles along rows (128/16)
  const int side = (wave >> 7) & 1;  // 0: (ind0,W1a)  1: (ind1,W1b)
  const int n    = wave >> 8;        // 32 groups

  const int*   ind = side ? ind1 : ind0;
  const float* W   = side ? W1b : W1a;

  // gathered A row for this lane's M index (lanes 0-15 carry M, per ISA layout)
  const int j = mt * 16 + lo;                       // row in [0,128)
  const int g = ind[n * LROWS + j] + BAG * (j & 3); // id in [0,1200)
  const float* Arow = orig + ((size_t)((n * 2 + side) * POOL + g)) * DDIM;

  const int h0 = ht * 16;
  v8f c = {};

  #pragma unroll 4
  for (int k0 = 0; k0 < DDIM; k0 += 4) {
    // A 16x4 f32: VGPR0/1 = K{0,1} (lanes 0-15) or K{2,3} (lanes 16-31)
    v2f a = *(const v2f*)(Arow + k0 + 2 * hi);   // 8B aligned (even k)
    // B 4x16 f32: mirrored layout, N across lanes, K split lo/hi halves
    v2f b;
    const int kb = (k0 + 2 * hi) * HDIM + h0 + lo;
    b.x = W[kb];
    b.y = W[kb + HDIM];
    c = __builtin_amdgcn_wmma_f32_16x16x4_f32(false, a, false, b,
                                              (short)0, c, false, false);
  }

  const float bias = side ? b1[h0 + lo] : 0.f;
  // C/D layout: VGPR v -> M = v + 8*hi, N = lo
  float* out = Hbuf + ((size_t)((n * 2 + side) * LROWS + mt * 16)) * HDIM + h0 + lo;
  #pragma unroll
  for (int v = 0; v < 8; ++v)
    out[(size_t)(v + 8 * hi) * HDIM] = c[v] + bias;
}

// ---------------------------------------------------------------------------
// Kernel 2: relu-relation score + 4x4 patch sum + pair indices.
// Block = (n, a-patch): 4 H0 rows in LDS, stream 128 H1 rows (L2-hot),
// 8 threads split the h-reduction per output, serial 8-way LDS sum at the end.
// ---------------------------------------------------------------------------
__global__ __launch_bounds__(256) void score_kernel(
    const float* __restrict__ Hbuf, const float* __restrict__ W2,
    const float* __restrict__ b2, float* __restrict__ scores,
    float* __restrict__ pairs)
{
  __shared__ float sH0[4 * HDIM];
  __shared__ float sW2[HDIM];
  __shared__ float sRed[256];

  const int n  = blockIdx.x >> 5;
  const int ap = blockIdx.x & 31;
  const int t  = threadIdx.x;

  const float* H0 = Hbuf + ((size_t)((n * 2 + 0) * LROWS) + ap * 4) * HDIM;
  const float* H1 = Hbuf + ((size_t)((n * 2 + 1) * LROWS)) * HDIM;

  for (int i = t; i < 4 * HDIM; i += 256) sH0[i] = H0[i];
  if (t < HDIM) sW2[t] = W2[t];
  __syncthreads();

  const int bp = t >> 3;   // 32 b-patches
  const int hc = t & 7;    // 8-way split of h
  const float* H1b = H1 + (size_t)(bp * 4) * HDIM;

  float acc = 0.f;
  for (int h = hc * 32; h < hc * 32 + 32; ++h) {
    const float w  = sW2[h];
    const float v0 = H1b[0 * HDIM + h];
    const float v1 = H1b[1 * HDIM + h];
    const float v2 = H1b[2 * HDIM + h];
    const float v3 = H1b[3 * HDIM + h];
    #pragma unroll
    for (int da = 0; da < 4; ++da) {
      const float u = sH0[da * HDIM + h];
      acc += fmaxf(u + v0, 0.f) * w;
      acc += fmaxf(u + v1, 0.f) * w;
      acc += fmaxf(u + v2, 0.f) * w;
      acc += fmaxf(u + v3, 0.f) * w;
    }
  }
  sRed[t] = acc;
  __syncthreads();

  if (t < 32) {
    float s = 0.f;
    #pragma unroll
    for (int i = 0; i < 8; ++i) s += sRed[t * 8 + i];
    const int idx = n * 1024 + ap * 32 + t;           // [num, m*l]
    scores[idx] = s + 16.f * b2[0];                   // 16 patch elems of +b2
    pairs[idx * 2 + 0] = (float)ap;                   // meshgrid 'ij'
    pairs[idx * 2 + 1] = (float)t;
  }
}

extern "C" void kernel_launch(void* const* d_in, const int* in_sizes, int n_in,
                              void* d_out, int out_size, void* d_ws, size_t ws_size,
                              hipStream_t stream) {
  const float* orig = (const float*)d_in[0];
  const int*   ind0 = (const int*)d_in[1];
  const int*   ind1 = (const int*)d_in[2];
  // d_in[3] = k (==8) is baked in as patch size 4
  const float* W1a  = (const float*)d_in[4];
  const float* W1b  = (const float*)d_in[5];
  const float* b1   = (const float*)d_in[6];
  const float* W2   = (const float*)d_in[7];
  const float* b2   = (const float*)d_in[8];

  float* Hbuf   = (float*)d_ws;                 // 32*2*128*256 f32 = 8 MB
  float* scores = (float*)d_out;                // 32*1024
  float* pairs  = (float*)d_out + 32 * 1024;    // 32*1024*2

  gemm_gather_wmma<<<1024, 256, 0, stream>>>(orig, ind0, ind1, W1a, W1b, b1, Hbuf);
  score_kernel<<<1024, 256, 0, stream>>>(Hbuf, W2, b2, scores, pairs);
}